// MotifInteractionGraph_83210696393638
// MI455X (gfx1250) — compile-verified
//
#include <hip/hip_runtime.h>
#include <hip/hip_bf16.h>

#define N_NODES 1024
#define HID     128

typedef float v2f __attribute__((ext_vector_type(2)));
typedef float v4f __attribute__((ext_vector_type(4)));
typedef float v8f __attribute__((ext_vector_type(8)));
typedef int   v4i __attribute__((ext_vector_type(4)));

#if defined(__gfx1250__) && __has_builtin(__builtin_amdgcn_wmma_f32_16x16x4_f32)
#define HAVE_WMMA_F32X4 1
#endif

// D = A(16x4, f32) * B(4x16, f32) + C(16x16, f32), wave32, exact fp32.
__device__ __forceinline__ v8f wmma_f32_16x16x4(v2f a, v2f b, v8f c) {
#ifdef HAVE_WMMA_F32X4
    return __builtin_amdgcn_wmma_f32_16x16x4_f32(false, a, false, b, (short)0, c,
                                                 false, false);
#else
    c[0] += a.x * b.x;
    c[1] += a.y * b.y;
    return c;
#endif
}

__device__ __forceinline__ unsigned ballot32(bool p) {
#if __has_builtin(__builtin_amdgcn_ballot_w32)
    return __builtin_amdgcn_ballot_w32(p);
#else
    return (unsigned)__ballot(p);
#endif
}

// Fragment loader shared by A and B operands of V_WMMA_F32_16X16X4_F32.
// A (16x4 MxK): lanes 0-15 hold A[lane, k+0..1], lanes 16-31 hold A[lane-16, k+2..3].
// B (4x16 KxN): lanes 0-15 hold B[k+0..1, lane], lanes 16-31 hold B[k+2..3, lane-16].
__device__ __forceinline__ v2f frag2(const float* __restrict__ p, int rowbase,
                                     int k, int lane) {
    const int r  = rowbase + (lane & 15);
    const int kk = k + ((lane >> 4) << 1);
    return *(const v2f*)&p[((long)r << 7) + kk];
}

// ---------------------------------------------------------------------------
// Kernel A: agg[i,h] = sum_j adj[i,j] * nf[j,h] * T[i,j,h]
// Block = node i; wave w owns contiguous edges j in [w*128, w*128+128).
// Step 1: ballot/popcount stream-compaction of live edge indices to LDS.
// Step 2: dense branch-free loop, 4 edges per iteration, 4 accumulators,
//         8 wide loads in flight per wave (NT for the 512MB T stream),
//         prefetch of the next index quad's rows.
// ---------------------------------------------------------------------------
__global__ void __launch_bounds__(256) agg_kernel(const float* __restrict__ nf,
                                                  const int* __restrict__ adj,
                                                  const float* __restrict__ T,
                                                  float* __restrict__ agg) {
    const int i    = blockIdx.x;
    const int tid  = threadIdx.x;
    const int wv   = tid >> 5;   // wave 0..7 -> contiguous 128-edge chunk
    const int lane = tid & 31;   // float4 group over H: h = 4*lane

    __shared__ __attribute__((aligned(16))) int list[8][128];
    __shared__ v4f red[8][32];

    const long rowbase = (long)i << 10;   // i*1024
    const int  jbase   = wv << 7;         // wave's first edge

    // ---- compact live edge indices (count ends up wave-uniform) ----------
    int cnt = 0;
    const unsigned lt = (1u << lane) - 1u;
#pragma unroll
    for (int c = 0; c < 4; ++c) {
        const int jloc = (c << 5) + lane;
        const int av   = adj[rowbase + jbase + jloc];
        const unsigned m = ballot32(av != 0);
        if (av) list[wv][cnt + __builtin_popcount(m & lt)] = jloc;
        cnt += __builtin_popcount(m);
    }
    __syncthreads();

    const float* __restrict__ Trow = T + ((rowbase + jbase) << 7);
    v4f a0 = {0.f, 0.f, 0.f, 0.f}, a1 = a0, a2 = a0, a3 = a0;
    const int hoff = lane << 2;

    int k = 0;
    for (; k + 4 <= cnt; k += 4) {
        const v4i q = *(const v4i*)&list[wv][k];   // uniform ds_load_b128

        if (k + 8 <= cnt) {                        // prefetch next quad's rows
            const v4i qn = *(const v4i*)&list[wv][k + 4];
            __builtin_prefetch(&Trow[((long)qn.x << 7) + hoff], 0, 1);
            __builtin_prefetch(&Trow[((long)qn.y << 7) + hoff], 0, 1);
            __builtin_prefetch(&Trow[((long)qn.z << 7) + hoff], 0, 1);
            __builtin_prefetch(&Trow[((long)qn.w << 7) + hoff], 0, 1);
        }

        const v4f t0 = __builtin_nontemporal_load((const v4f*)&Trow[((long)q.x << 7) + hoff]);
        const v4f t1 = __builtin_nontemporal_load((const v4f*)&Trow[((long)q.y << 7) + hoff]);
        const v4f t2 = __builtin_nontemporal_load((const v4f*)&Trow[((long)q.z << 7) + hoff]);
        const v4f t3 = __builtin_nontemporal_load((const v4f*)&Trow[((long)q.w << 7) + hoff]);
        const v4f f0 = *(const v4f*)&nf[((long)(jbase + q.x) << 7) + hoff];
        const v4f f1 = *(const v4f*)&nf[((long)(jbase + q.y) << 7) + hoff];
        const v4f f2 = *(const v4f*)&nf[((long)(jbase + q.z) << 7) + hoff];
        const v4f f3 = *(const v4f*)&nf[((long)(jbase + q.w) << 7) + hoff];

        a0 += f0 * t0;
        a1 += f1 * t1;
        a2 += f2 * t2;
        a3 += f3 * t3;
    }
    for (; k < cnt; ++k) {                         // <=3 leftover edges
        const int j = list[wv][k];
        const v4f t = __builtin_nontemporal_load((const v4f*)&Trow[((long)j << 7) + hoff]);
        const v4f f = *(const v4f*)&nf[((long)(jbase + j) << 7) + hoff];
        a0 += f * t;
    }

    red[wv][lane] = (a0 + a1) + (a2 + a3);
    __syncthreads();

    if (tid < 32) {
        v4f s = red[0][lane];
#pragma unroll
        for (int g = 1; g < 8; ++g) s += red[g][lane];
        *(v4f*)&agg[((long)i << 7) + (lane << 2)] = s;
    }
}

// ---------------------------------------------------------------------------
// Kernel B: GRUCell via fp32 WMMA (unchanged from round 1 — compiled to
// dual v_wmma_f32_16x16x4_f32 chains with staged loadcnt pipelining).
// ---------------------------------------------------------------------------
__global__ void __launch_bounds__(256) gru_kernel(const float* __restrict__ agg,
                                                  const float* __restrict__ nf,
                                                  const float* __restrict__ w_ih,
                                                  const float* __restrict__ w_hh,
                                                  const float* __restrict__ b_ih,
                                                  const float* __restrict__ b_hh,
                                                  float* __restrict__ out) {
    __shared__ float Grz[16][264];  // r|z pre-activations (gi+gh), padded
    __shared__ float Gni[16][132];  // n-gate input part  (gi)
    __shared__ float Gnh[16][132];  // n-gate hidden part (gh)

    const int n0    = blockIdx.x << 4;
    const int tid   = threadIdx.x;
    const int wave  = tid >> 5;
    const int lane  = tid & 31;
    const int col   = lane & 15;
    const int rbase = (lane >> 4) << 3;  // C/D: VGPR r -> row r / r+8

    // ---------------- phase 1: r and z gates (shared A fragments) ----------
    {
        const int m0r = wave << 4;        // columns   0..127  (r)
        const int m0z = (wave + 8) << 4;  // columns 128..255  (z)
        v8f accr, accz;
        const float br_ = b_ih[m0r + col] + b_hh[m0r + col];
        const float bz_ = b_ih[m0z + col] + b_hh[m0z + col];
#pragma unroll
        for (int r = 0; r < 8; ++r) { accr[r] = br_; accz[r] = bz_; }

#pragma unroll 4
        for (int k = 0; k < HID; k += 4) {  // gi contribution
            const v2f a  = frag2(agg, n0, k, lane);
            const v2f wr = frag2(w_ih, m0r, k, lane);
            const v2f wz = frag2(w_ih, m0z, k, lane);
            accr = wmma_f32_16x16x4(a, wr, accr);
            accz = wmma_f32_16x16x4(a, wz, accz);
        }
#pragma unroll 4
        for (int k = 0; k < HID; k += 4) {  // gh contribution
            const v2f h  = frag2(nf, n0, k, lane);
            const v2f wr = frag2(w_hh, m0r, k, lane);
            const v2f wz = frag2(w_hh, m0z, k, lane);
            accr = wmma_f32_16x16x4(h, wr, accr);
            accz = wmma_f32_16x16x4(h, wz, accz);
        }
#pragma unroll
        for (int r = 0; r < 8; ++r) {
            Grz[rbase + r][m0r + col] = accr[r];
            Grz[rbase + r][m0z + col] = accz[r];
        }
    }

    // ---------------- phase 2: n gate (gi, gh kept separate) ---------------
    {
        const int m0n  = (16 + wave) << 4;  // columns 256..383
        const int mloc = m0n - 256;
        v8f acci, acch;
        const float bi_ = b_ih[m0n + col];
        const float bh_ = b_hh[m0n + col];
#pragma unroll
        for (int r = 0; r < 8; ++r) { acci[r] = bi_; acch[r] = bh_; }

#pragma unroll 4
        for (int k = 0; k < HID; k += 4) {
            const v2f ai = frag2(agg, n0, k, lane);
            const v2f ah = frag2(nf,  n0, k, lane);
            const v2f wi = frag2(w_ih, m0n, k, lane);
            const v2f wh = frag2(w_hh, m0n, k, lane);
            acci = wmma_f32_16x16x4(ai, wi, acci);
            acch = wmma_f32_16x16x4(ah, wh, acch);
        }
#pragma unroll
        for (int r = 0; r < 8; ++r) {
            Gni[rbase + r][mloc + col] = acci[r];
            Gnh[rbase + r][mloc + col] = acch[r];
        }
    }

    __syncthreads();

    // ---------------- fused elementwise GRU epilogue -----------------------
#pragma unroll
    for (int e = 0; e < 8; ++e) {
        const int idx = tid + (e << 8);  // 0..2047
        const int row = idx >> 7;
        const int h   = idx & 127;
        const float rg = 1.f / (1.f + __expf(-Grz[row][h]));
        const float zg = 1.f / (1.f + __expf(-Grz[row][128 + h]));
        const float ng = tanhf(Gni[row][h] + rg * Gnh[row][h]);
        const long  o  = ((long)(n0 + row) << 7) + h;
        out[o] = (1.f - zg) * ng + zg * nf[o];
    }
}

// ---------------------------------------------------------------------------
extern "C" void kernel_launch(void* const* d_in, const int* in_sizes, int n_in,
                              void* d_out, int out_size, void* d_ws, size_t ws_size,
                              hipStream_t stream) {
    const float* nf   = (const float*)d_in[0];  // [1024,128]
    const int*   adj  = (const int*)  d_in[1];  // [1024,1024]
    const float* T    = (const float*)d_in[2];  // [1024*1024,128]
    const float* w_ih = (const float*)d_in[3];  // [384,128]
    const float* w_hh = (const float*)d_in[4];  // [384,128]
    const float* b_ih = (const float*)d_in[5];  // [384]
    const float* b_hh = (const float*)d_in[6];  // [384]
    float* out = (float*)d_out;
    float* agg = (float*)d_ws;                  // 1024*128 f32 = 512 KB scratch

    agg_kernel<<<dim3(N_NODES), dim3(256), 0, stream>>>(nf, adj, T, agg);
    gru_kernel<<<dim3(N_NODES / 16), dim3(256), 0, stream>>>(agg, nf, w_ih, w_hh,
                                                             b_ih, b_hh, out);
}